// GATLayer_81381040325136
// MI455X (gfx1250) — compile-verified
//
#include <hip/hip_runtime.h>
#include <hip/hip_bf16.h>

typedef float v2f __attribute__((ext_vector_type(2)));
typedef float v8f __attribute__((ext_vector_type(8)));

#define N_NODES 100000
#define N_EDGES 1600000
#define IN_C    128
#define OUT_C   32
#define HEADS   4
#define FEAT    (HEADS * OUT_C)   // 128
#define LDS_STRIDE 132            // 128 + 4 pad -> conflict-free column reads

// monotone float <-> uint encoding so atomicMax(uint) == float max
__device__ __forceinline__ unsigned enc_f32(float f) {
    unsigned u = __float_as_uint(f);
    return (u & 0x80000000u) ? ~u : (u | 0x80000000u);
}
__device__ __forceinline__ float dec_f32(unsigned u) {
    u = (u & 0x80000000u) ? (u & 0x7FFFFFFFu) : ~u;
    return __uint_as_float(u);
}

// ---------------------------------------------------------------------------
// Kernel 1: h[n, head*32+o] = sum_i x[n,i] * W[head,i,o]
// grid = 6250 (16-node tiles), block = 256 (8 waves; wave w owns cols 16w..16w+15)
// V_WMMA_F32_16X16X4_F32: A = 16x4 (2 VGPR), B = 4x16 (2 VGPR), C/D = 16x16 (8 VGPR)
// ---------------------------------------------------------------------------
__global__ __launch_bounds__(256) void gat_transform(
        const float* __restrict__ x, const float* __restrict__ W,
        float* __restrict__ h) {
    __shared__ float xs[16 * LDS_STRIDE];
    const int m = blockIdx.x;
    const int t = threadIdx.x;

    // stage 16x128 tile of x into LDS (each thread: 8 contiguous floats)
    {
        const int row = t >> 4;
        const int c0  = (t & 15) * 8;
        const float4* src = (const float4*)(x + (size_t)(m * 16 + row) * IN_C + c0);
        *(float4*)(xs + row * LDS_STRIDE + c0)     = src[0];
        *(float4*)(xs + row * LDS_STRIDE + c0 + 4) = src[1];
    }
    __syncthreads();

    const int wave = t >> 5;          // N tile 0..7
    const int lane = t & 31;
    const int half = lane >> 4;       // 0: K-offsets {0,1}; 1: K-offsets {2,3}
    const int idx  = lane & 15;       // row (A) / col (B) within tile
    const int koff = half * 2;

    const int col  = wave * 16 + idx; // global output column 0..127
    const int head = col >> 5;
    const int oc   = col & 31;

    // preload all B fragments for this column (W is 64 KB, L2-resident)
    v2f b[32];
#pragma unroll
    for (int kk = 0; kk < 32; ++kk) {
        const int k = kk * 4 + koff;
        b[kk].x = W[head * (IN_C * OUT_C) + (k    ) * OUT_C + oc];
        b[kk].y = W[head * (IN_C * OUT_C) + (k + 1) * OUT_C + oc];
    }

    v8f c = {};
#pragma unroll
    for (int kk = 0; kk < 32; ++kk) {
        const int k = kk * 4 + koff;
        v2f a;
        a.x = xs[idx * LDS_STRIDE + k];
        a.y = xs[idx * LDS_STRIDE + k + 1];
        c = __builtin_amdgcn_wmma_f32_16x16x4_f32(false, a, false, b[kk],
                                                  (short)0, c, false, false);
    }

    // D layout: VGPR r -> (M=r, N=lane) lanes 0..15 ; (M=r+8, N=lane-16) lanes 16..31
#pragma unroll
    for (int r = 0; r < 8; ++r) {
        const int node = m * 16 + half * 8 + r;
        h[(size_t)node * FEAT + col] = c[r];
    }
}

// ---------------------------------------------------------------------------
// Kernel 2: per-node attention scalars el/er  (thread per (node, head))
// ---------------------------------------------------------------------------
__global__ __launch_bounds__(256) void gat_alpha(
        const float* __restrict__ h, const float* __restrict__ a,
        float* __restrict__ el, float* __restrict__ er) {
    const int t = blockIdx.x * blockDim.x + threadIdx.x;
    if (t >= N_NODES * HEADS) return;
    const int n  = t >> 2;
    const int hd = t & 3;
    const float* hp = h + (size_t)n * FEAT + hd * OUT_C;
    const float* as = a + hd * 2 * OUT_C;
    const float* ad = as + OUT_C;
    float s = 0.f, d = 0.f;
#pragma unroll
    for (int o = 0; o < OUT_C; o += 4) {
        float4 hv = *(const float4*)(hp + o);
        float4 sv = *(const float4*)(as + o);
        float4 dv = *(const float4*)(ad + o);
        s += hv.x * sv.x + hv.y * sv.y + hv.z * sv.z + hv.w * sv.w;
        d += hv.x * dv.x + hv.y * dv.y + hv.z * dv.z + hv.w * dv.w;
    }
    el[t] = s;
    er[t] = d;
}

// ---------------------------------------------------------------------------
// Kernel 3: reset accumulators (graph-replay safe)
// ---------------------------------------------------------------------------
__global__ __launch_bounds__(256) void gat_init(
        unsigned* __restrict__ dmax, float* __restrict__ dsum,
        float* __restrict__ out) {
    const int t = blockIdx.x * blockDim.x + threadIdx.x;
    if (t < N_NODES * HEADS) { dmax[t] = 0u; dsum[t] = 0.f; }
    if (t < N_NODES * FEAT)  { out[t]  = 0.f; }
}

// ---------------------------------------------------------------------------
// Kernel 4: segment max of leaky-relu logits  (thread per (edge, head))
// ---------------------------------------------------------------------------
__global__ __launch_bounds__(256) void gat_edge_max(
        const long long* __restrict__ ei,
        const float* __restrict__ el, const float* __restrict__ er,
        unsigned* __restrict__ dmax) {
    const int t = blockIdx.x * blockDim.x + threadIdx.x;
    if (t >= N_EDGES * HEADS) return;
    const int e  = t >> 2;
    const int hd = t & 3;
    const int s = (int)ei[e];
    const int d = (int)ei[N_EDGES + e];
    float v = el[s * HEADS + hd] + er[d * HEADS + hd];
    v = v > 0.f ? v : 0.2f * v;
    atomicMax(&dmax[d * HEADS + hd], enc_f32(v));
}

// ---------------------------------------------------------------------------
// Kernel 5: segment sum of exp(e - max)  (thread per (edge, head))
// ---------------------------------------------------------------------------
__global__ __launch_bounds__(256) void gat_edge_sum(
        const long long* __restrict__ ei,
        const float* __restrict__ el, const float* __restrict__ er,
        const unsigned* __restrict__ dmax, float* __restrict__ dsum) {
    const int t = blockIdx.x * blockDim.x + threadIdx.x;
    if (t >= N_EDGES * HEADS) return;
    const int e  = t >> 2;
    const int hd = t & 3;
    const int s = (int)ei[e];
    const int d = (int)ei[N_EDGES + e];
    float v = el[s * HEADS + hd] + er[d * HEADS + hd];
    v = v > 0.f ? v : 0.2f * v;
    atomicAdd(&dsum[d * HEADS + hd], __expf(v - dec_f32(dmax[d * HEADS + hd])));
}

// ---------------------------------------------------------------------------
// Kernel 6: out[dst] += alpha * h[src]   (one wave per edge, 4 channels/lane)
// h and out are L2-resident (51.2 MB each << 192 MB) -> atomics resolve in L2
// ---------------------------------------------------------------------------
__global__ __launch_bounds__(256) void gat_aggregate(
        const long long* __restrict__ ei,
        const float* __restrict__ el, const float* __restrict__ er,
        const unsigned* __restrict__ dmax, const float* __restrict__ dsum,
        const float* __restrict__ h, float* __restrict__ out) {
    const int t    = blockIdx.x * blockDim.x + threadIdx.x;
    const int e    = t >> 5;
    const int lane = t & 31;
    if (e >= N_EDGES) return;
    const int hd = lane >> 3;                 // head for this lane's 4 channels
    const int s = (int)ei[e];
    const int d = (int)ei[N_EDGES + e];
    float v = el[s * HEADS + hd] + er[d * HEADS + hd];
    v = v > 0.f ? v : 0.2f * v;
    const float coef = __expf(v - dec_f32(dmax[d * HEADS + hd]))
                     / dsum[d * HEADS + hd];
    const float4 hv = *(const float4*)(h + (size_t)s * FEAT + lane * 4);
    float* op = out + (size_t)d * FEAT + lane * 4;
    atomicAdd(op + 0, coef * hv.x);
    atomicAdd(op + 1, coef * hv.y);
    atomicAdd(op + 2, coef * hv.z);
    atomicAdd(op + 3, coef * hv.w);
}

// ---------------------------------------------------------------------------
extern "C" void kernel_launch(void* const* d_in, const int* in_sizes, int n_in,
                              void* d_out, int out_size, void* d_ws, size_t ws_size,
                              hipStream_t stream) {
    const float*     x  = (const float*)d_in[0];
    const long long* ei = (const long long*)d_in[1];   // int64 per reference
    const float*     W  = (const float*)d_in[2];
    const float*     a  = (const float*)d_in[3];
    float* out = (float*)d_out;

    // workspace layout (~57.7 MB)
    float*    h    = (float*)d_ws;                          // N*128
    float*    el   = h + (size_t)N_NODES * FEAT;            // N*4
    float*    er   = el + (size_t)N_NODES * HEADS;          // N*4
    unsigned* dmax = (unsigned*)(er + (size_t)N_NODES * HEADS); // N*4
    float*    dsum = (float*)(dmax + (size_t)N_NODES * HEADS);  // N*4

    const int B = 256;
    gat_init<<<(N_NODES * FEAT + B - 1) / B, B, 0, stream>>>(dmax, dsum, out);
    gat_transform<<<N_NODES / 16, B, 0, stream>>>(x, W, h);
    gat_alpha<<<(N_NODES * HEADS + B - 1) / B, B, 0, stream>>>(h, a, el, er);
    gat_edge_max<<<(N_EDGES * HEADS + B - 1) / B, B, 0, stream>>>(ei, el, er, dmax);
    gat_edge_sum<<<(N_EDGES * HEADS + B - 1) / B, B, 0, stream>>>(ei, el, er, dmax, dsum);
    const long agg_threads = (long)N_EDGES * 32;
    gat_aggregate<<<(int)((agg_threads + B - 1) / B), B, 0, stream>>>(
        ei, el, er, dmax, dsum, h, out);
}